// SpatialGaussianBlur_86947317940565
// MI455X (gfx1250) — compile-verified
//
#include <hip/hip_runtime.h>
#include <cstdint>

// 5x5 separable Gaussian blur (sigma=1, reflect pad) on (t*z, 512, 512) f32.
// HBM-bound (~1 GiB traffic): single fused pass through LDS,
// async b128 global->LDS staging (GVS form), vectorized LDS traffic,
// nontemporal b128 output stores.

#define WIDTH      512
#define HEIGHT     512
#define TX         128          // output tile width
#define TY         32           // output tile height
#define INH        36           // staged rows  (TY + 4)
#define SINW       136          // staged cols: gx in [x0-4, x0+132), 16B aligned
#define NTHREADS   256

typedef float v2f __attribute__((ext_vector_type(2)));
typedef float v4f __attribute__((ext_vector_type(4)));

__device__ __forceinline__ int reflect512(int i) {
    // reflect (no edge dup): -1->1, -2->2, 512->510, 513->509.
    // = min(max(i, -i), 1022 - i), valid for i in (-512, 1022).
    const int a = (i < 0) ? -i : i;
    const int b = 1022 - i;
    return (a < b) ? a : b;
}

__global__ __launch_bounds__(NTHREADS)
void gauss5_blur_kernel(const float* __restrict__ src, float* __restrict__ dst) {
    __shared__ alignas(16) float s_in[INH][SINW]; // staged reflect-padded tile
    __shared__ alignas(16) float s_h[INH][TX];    // horizontal-pass result

    const int    tid    = threadIdx.x;
    const int    lane_g = tid & 31;               // column group 0..31
    const int    wv     = tid >> 5;               // wave id 0..7
    const int    c4     = lane_g << 2;            // 0,4,...,124
    const int    x0     = blockIdx.x * TX;
    const int    y0     = blockIdx.y * TY;
    const size_t slice  = (size_t)blockIdx.z * (size_t)(WIDTH * HEIGHT);

    // Scalar base = &src[slice + x0 - 4]  (column 0 of the staged tile)
    const uint64_t base = (uint64_t)(uintptr_t)(src + slice + (size_t)x0) - 16u;

    // ---- Phase 1a: interior staging, one b128 async load per (row, group) --
    // s_in[r][4 + (gx-x0)] layout; groups cover gx in [x0, x0+128).
    {
        const int goff = lane_g << 4;             // byte offset of group in row
        for (int r = wv; r < INH; r += 8) {
            const int gy   = reflect512(y0 + r - 2);
            const int voff = (gy << 11) + 16 + goff;         // gy*2048 + (4+4g)*4
            const uint32_t loff = (uint32_t)(uintptr_t)&s_in[r][4 + c4];
            asm volatile("global_load_async_to_lds_b128 %0, %1, %2"
                         :: "v"(loff), "v"(voff), "s"(base)
                         : "memory");
        }
    }
    // ---- Phase 1b: halo columns gx = x0-2, x0-1, x0+128, x0+129 (reflected) -
    if (tid < 4 * INH) {
        const int r    = tid >> 2;
        const int wcol = tid & 3;
        const int idx  = (wcol < 2) ? (2 + wcol) : (130 + wcol); // 2,3,132,133
        const int gx   = reflect512(x0 - 4 + idx);
        const int gy   = reflect512(y0 + r - 2);
        const int voff = (gy << 11) + ((gx - x0 + 4) << 2);
        const uint32_t loff = (uint32_t)(uintptr_t)&s_in[r][idx];
        asm volatile("global_load_async_to_lds_b32 %0, %1, %2"
                     :: "v"(loff), "v"(voff), "s"(base)
                     : "memory");
    }

#if __has_builtin(__builtin_amdgcn_s_wait_asynccnt)
    __builtin_amdgcn_s_wait_asynccnt(0);
#else
    asm volatile("s_wait_asynccnt 0" ::: "memory");
#endif
    __syncthreads();

    // Normalized 5-tap Gaussian, sigma = 1 (matches reference)
    const float KW0 = 0.05448868454964294f;
    const float KW1 = 0.24420134200323332f;
    const float KW2 = 0.40261994689402953f;

    // ---- Phase 2: horizontal pass, v4 per thread-task -----------------------
    // out col c -> s_in idx c+2 .. c+6 ; window idx c4+2 .. c4+9 (8 floats).
    for (int r = wv; r < INH; r += 8) {
        const v2f lo  = *reinterpret_cast<const v2f*>(&s_in[r][c4 + 2]); // 8B aligned
        const v4f mid = *reinterpret_cast<const v4f*>(&s_in[r][c4 + 4]); // 16B aligned
        const v2f hi  = *reinterpret_cast<const v2f*>(&s_in[r][c4 + 8]); // 8B aligned
        const float w0 = lo.x,  w1 = lo.y,  w2 = mid.x, w3 = mid.y;
        const float w4 = mid.z, w5 = mid.w, w6 = hi.x,  w7 = hi.y;
        v4f o;
        o.x = KW0 * (w0 + w4) + KW1 * (w1 + w3) + KW2 * w2;
        o.y = KW0 * (w1 + w5) + KW1 * (w2 + w4) + KW2 * w3;
        o.z = KW0 * (w2 + w6) + KW1 * (w3 + w5) + KW2 * w4;
        o.w = KW0 * (w3 + w7) + KW1 * (w4 + w6) + KW2 * w5;
        *reinterpret_cast<v4f*>(&s_h[r][c4]) = o;
    }
    __syncthreads();

    // ---- Phase 3: vertical pass, 4 consecutive rows per thread --------------
    const int r4 = wv << 2;                        // out rows r4 .. r4+3
    float* op = dst + slice + (size_t)(y0 + r4) * WIDTH + x0 + c4;
    v4f h[8];
    #pragma unroll
    for (int k = 0; k < 8; ++k)
        h[k] = *reinterpret_cast<const v4f*>(&s_h[r4 + k][c4]);
    #pragma unroll
    for (int j = 0; j < 4; ++j) {
        const v4f acc = KW0 * (h[j] + h[j + 4])
                      + KW1 * (h[j + 1] + h[j + 3])
                      + KW2 * h[j + 2];
        __builtin_nontemporal_store(acc, reinterpret_cast<v4f*>(op));
        op += WIDTH;
    }
}

extern "C" void kernel_launch(void* const* d_in, const int* in_sizes, int n_in,
                              void* d_out, int out_size, void* d_ws, size_t ws_size,
                              hipStream_t stream) {
    (void)n_in; (void)out_size; (void)d_ws; (void)ws_size;
    const float* x = (const float*)d_in[0];
    float*       y = (float*)d_out;
    const int slices = in_sizes[0] / (WIDTH * HEIGHT);   // t*z = 512
    dim3 grid(WIDTH / TX, HEIGHT / TY, slices);          // (4, 16, 512)
    gauss5_blur_kernel<<<grid, dim3(NTHREADS), 0, stream>>>(x, y);
}